// ResiduesEncodeV2_51719996178661
// MI455X (gfx1250) — compile-verified
//
#include <hip/hip_runtime.h>

typedef __attribute__((ext_vector_type(2))) float v2f;
typedef __attribute__((ext_vector_type(8))) float v8f;

#define BATCH 32
#define RTOT  4000
#define RLET  800
#define TA    49600
#define VDIM  64

#define RES_STRIDE 208          // LDS floats per residue (192 data + pad)
#define WAVE_LDSF  (16 * RES_STRIDE)

// ---------------------------------------------------------------------------
// Prep: collapse W_enc (64, n*n) into Weff (64, Kpad) with zero K-padding,
// and build center weights wc[a] = 1/n + W_ctr[a], scalar S = sum(W_ctr).
// ---------------------------------------------------------------------------
__global__ void prep_kernel(const float* __restrict__ Wenc,
                            const float* __restrict__ Wctr,
                            float* __restrict__ weff,   // [64][Kpad]
                            float* __restrict__ wc,     // [n]
                            float* __restrict__ Sout,   // [1]
                            int n, int Kpad) {
  int v = threadIdx.x;  // 0..63
  for (int a = 0; a < Kpad; ++a) {
    float val = 0.0f;
    if (a < n) {
      float cs = 0.0f, rs = 0.0f;
      for (int p = 0; p < n; ++p) {
        cs += Wenc[v * n * n + p * n + a];
        rs += Wenc[v * n * n + a * n + p];
      }
      val = cs - rs;
    }
    weff[v * Kpad + a] = val;
  }
  if (v < n) wc[v] = 1.0f / (float)n + Wctr[v];
  if (v == 0) {
    float s = 0.0f;
    for (int a = 0; a < n; ++a) s += Wctr[a];
    Sout[0] = s;
  }
}

// ---------------------------------------------------------------------------
// Centers: one thread per (b, r) of a letter. 3 output floats each.
// ---------------------------------------------------------------------------
template <int N>
__global__ void ctr_kernel(const float* __restrict__ x,
                           const float* __restrict__ wc,
                           const float* __restrict__ Sptr,
                           const int* __restrict__ idx,
                           const int* __restrict__ pos,
                           float* __restrict__ ctr_out) {
  int t = blockIdx.x * blockDim.x + threadIdx.x;
  if (t >= BATCH * RLET) return;
  int b = t / RLET;
  int r = t - b * RLET;
  int a0 = idx[r * N];
  const float* xp = x + ((size_t)b * TA + (size_t)a0) * 3;
  float s0 = 0.0f, s1 = 0.0f, s2 = 0.0f;
#pragma unroll
  for (int a = 0; a < N; ++a) {
    float w = wc[a];
    s0 += w * xp[a * 3 + 0];
    s1 += w * xp[a * 3 + 1];
    s2 += w * xp[a * 3 + 2];
  }
  float S = Sptr[0];
  float* o = ctr_out + ((size_t)b * RTOT + (size_t)pos[r]) * 3;
  o[0] = s0 - S;
  o[1] = s1 - S;
  o[2] = s2 - S;
}

// ---------------------------------------------------------------------------
// Encodings via V_WMMA_F32_16X16X4_F32, fully unrolled per letter.
// One wave = 16 residues of one (letter, batch): D(64 x 48) = Weff(64 x KPAD)
// * X(KPAD x 48), columns c -> (r = r0 + c/3, d = c%3).
// Full K-steps use base+immediate-offset loads (no per-step address math);
// at most one tail step per letter resolves clamps at compile time per hi.
// Epilogue: acc -> LDS (residue-major transpose), then
// GLOBAL_STORE_ASYNC_FROM_LDS_B128 drains 768B-contiguous residue blocks.
// ---------------------------------------------------------------------------
template <int N, int KPAD>
__global__ void __launch_bounds__(128)
enc_wmma_kernel(const float* __restrict__ x,
                const float* __restrict__ weff,
                const int* __restrict__ idx,
                const int* __restrict__ pos,
                float* __restrict__ enc_out) {
  __shared__ float smem[4 * WAVE_LDSF];   // 52 KB: one 16x208 tile per wave

  const int lane = threadIdx.x & 31;
  const int wv   = threadIdx.x >> 5;
  const int g    = blockIdx.x * 4 + wv;
  if (g >= RLET / 16) return;             // uniform per wave: EXEC all-1s
  const int b   = blockIdx.y;
  const int r0  = g * 16;
  const int l16 = lane & 15;
  const int hi  = lane >> 4;

  // Per-N-tile, per-lane bases (column c -> residue rl = c/3, dim d = c%3).
  const float* xp[3];                     // incl. d and 2*hi K-striping (6*hi floats)
  const float* xq[3];                     // incl. d only (for clamped tail step)
  int lbase[3];                           // LDS float index: rl*208 + d
#pragma unroll
  for (int t = 0; t < 3; ++t) {
    int c  = t * 16 + l16;                // 0..47
    int rl = c / 3;
    int d  = c - rl * 3;
    int a0 = idx[(r0 + rl) * N];          // atoms contiguous per residue
    const float* base = x + ((size_t)b * TA + (size_t)a0) * 3 + d;
    xq[t] = base;
    xp[t] = base + 6 * hi;
    lbase[t] = rl * RES_STRIDE + d;
  }

  const float* aw = weff + l16 * KPAD + 2 * hi;   // per-lane A base

  v8f acc[4][3];
#pragma unroll
  for (int m = 0; m < 4; ++m)
#pragma unroll
    for (int t = 0; t < 3; ++t)
      acc[m][t] = (v8f){};

  constexpr int KSTEPS = KPAD / 4;
#pragma unroll
  for (int ks = 0; ks < KSTEPS; ++ks) {
    // A fragments: immediate offsets off the per-lane base (zero-padded past N).
    v2f Af[4];
#pragma unroll
    for (int m = 0; m < 4; ++m) {
      const float* ap = aw + m * 16 * KPAD + ks * 4;
      Af[m].x = ap[0];
      Af[m].y = ap[1];
    }

    // B fragments.
    v2f Bf[3];
    if (ks * 4 + 3 < N) {
      // Full step: pure immediate-offset loads.
#pragma unroll
      for (int t = 0; t < 3; ++t) {
        Bf[t].x = xp[t][ks * 12];
        Bf[t].y = xp[t][ks * 12 + 3];
      }
    } else {
      // Tail step: clamp K past N (A rows there are zero, product dies).
      int k0 = ks * 4 + 2 * hi;     if (k0 > N - 1) k0 = N - 1;
      int k1 = ks * 4 + 1 + 2 * hi; if (k1 > N - 1) k1 = N - 1;
#pragma unroll
      for (int t = 0; t < 3; ++t) {
        Bf[t].x = xq[t][k0 * 3];
        Bf[t].y = xq[t][k1 * 3];
      }
    }

#pragma unroll
    for (int m = 0; m < 4; ++m)
#pragma unroll
      for (int t = 0; t < 3; ++t)
        acc[m][t] = __builtin_amdgcn_wmma_f32_16x16x4_f32(
            false, Af[m], false, Bf[t], (short)0, acc[m][t], false, false);
  }

  // --- Epilogue phase 1: transpose acc into this wave's LDS tile ---------
  float* my = smem + wv * WAVE_LDSF;      // [16][RES_STRIDE], data in [0,192)
#pragma unroll
  for (int m = 0; m < 4; ++m)
#pragma unroll
    for (int t = 0; t < 3; ++t) {
      float* lp = my + lbase[t];
#pragma unroll
      for (int j = 0; j < 8; ++j) {
        int v = m * 16 + j + 8 * hi;
        lp[v * 3] = acc[m][t][j];         // idx = rl*208 + v*3 + d
      }
    }

  // All DS writes from this wave must land before the async engine reads LDS.
  asm volatile("s_wait_dscnt 0x0" ::: "memory");

  // --- Epilogue phase 2: async drain, 768B contiguous per residue --------
  // 16 residues x 192 floats = 3072 floats = 24 iters x (32 lanes x 16B).
#pragma unroll
  for (int i = 0; i < 24; ++i) {
    int f = i * 128 + lane * 4;           // flat float index in tile
    int r = f / 192;                      // local residue
    int e = f - r * 192;                  // float offset within residue
    unsigned lds_b = (unsigned)(size_t)(my + r * RES_STRIDE + e);  // LDS byte addr
    float* gp = enc_out + ((size_t)b * RTOT + (size_t)pos[r0 + r]) * (VDIM * 3) + e;
    asm volatile("global_store_async_from_lds_b128 %0, %1, off"
                 :: "v"(gp), "v"(lds_b)
                 : "memory");
  }
  asm volatile("s_wait_asynccnt 0x0" ::: "memory");
}

// ---------------------------------------------------------------------------
template <int N, int KPAD>
static void launch_letter(const float* x, float* ws,
                          int weff_off, int wc_off, int s_off_l,
                          const float* Wenc, const float* Wctr,
                          const int* idx, const int* pos,
                          float* ctr_out, float* enc_out, hipStream_t stream) {
  prep_kernel<<<1, 64, 0, stream>>>(Wenc, Wctr, ws + weff_off, ws + wc_off,
                                    ws + s_off_l, N, KPAD);
  const int nthr = BATCH * RLET;
  ctr_kernel<N><<<(nthr + 255) / 256, 256, 0, stream>>>(
      x, ws + wc_off, ws + s_off_l, idx, pos, ctr_out);
  // 50 groups of 16 residues per batch; 4 waves per 128-thread block.
  enc_wmma_kernel<N, KPAD><<<dim3(13, BATCH), 128, 0, stream>>>(
      x, ws + weff_off, idx, pos, enc_out);
}

extern "C" void kernel_launch(void* const* d_in, const int* in_sizes, int n_in,
                              void* d_out, int out_size, void* d_ws, size_t ws_size,
                              hipStream_t stream) {
  (void)in_sizes; (void)n_in; (void)out_size; (void)ws_size;

  static const int ns[5]    = {7, 10, 12, 14, 19};
  static const int kpads[5] = {8, 12, 12, 16, 20};

  int weff_off[5], wc_off[5];
  int o = 0;
  for (int l = 0; l < 5; ++l) { weff_off[l] = o; o += VDIM * kpads[l]; }
  for (int l = 0; l < 5; ++l) { wc_off[l]   = o; o += ns[l]; }
  const int s_off = o;  // 5 scalars; ~4.4K floats of workspace total

  const float* x = (const float*)d_in[0];
  float* ws      = (float*)d_ws;
  float* ctr_out = (float*)d_out;
  float* enc_out = (float*)d_out + (size_t)BATCH * RTOT * 3;

  for (int l = 0; l < 5; ++l) {
    const float* Wenc = (const float*)d_in[1 + 4 * l];
    const float* Wctr = (const float*)d_in[2 + 4 * l];
    const int*   idx  = (const int*)  d_in[3 + 4 * l];
    const int*   pos  = (const int*)  d_in[4 + 4 * l];

    switch (l) {
      case 0: launch_letter<7, 8>(x, ws, weff_off[l], wc_off[l], s_off + l,
                                  Wenc, Wctr, idx, pos, ctr_out, enc_out, stream); break;
      case 1: launch_letter<10, 12>(x, ws, weff_off[l], wc_off[l], s_off + l,
                                    Wenc, Wctr, idx, pos, ctr_out, enc_out, stream); break;
      case 2: launch_letter<12, 12>(x, ws, weff_off[l], wc_off[l], s_off + l,
                                    Wenc, Wctr, idx, pos, ctr_out, enc_out, stream); break;
      case 3: launch_letter<14, 16>(x, ws, weff_off[l], wc_off[l], s_off + l,
                                    Wenc, Wctr, idx, pos, ctr_out, enc_out, stream); break;
      case 4: launch_letter<19, 20>(x, ws, weff_off[l], wc_off[l], s_off + l,
                                    Wenc, Wctr, idx, pos, ctr_out, enc_out, stream); break;
    }
  }
}